// GNN_72713796321966
// MI455X (gfx1250) — compile-verified
//
#include <hip/hip_runtime.h>
#include <hip/hip_bf16.h>

typedef __attribute__((ext_vector_type(2))) float v2f;
typedef __attribute__((ext_vector_type(8))) float v8f;

#define NFEAT 128
#define XS_STRIDE 132   // 128 + 4 pad: 16B-aligned rows, conflict-free ds reads

// ---------------------------------------------------------------------------
// Dense node-feature GEMM: H[n,128] = act(X[n,128]) @ W[128,128]
// One block = 16 rows; 8 waves, each computes one 16x16 tile via
// 32 chained V_WMMA_F32_16X16X4_F32 (K=128 in steps of 4).
// ---------------------------------------------------------------------------
__global__ __launch_bounds__(256) void gat_gemm128(
    const float* __restrict__ X, const float* __restrict__ W,
    float* __restrict__ H, int relu_in)
{
    __shared__ __align__(16) float Xs[16 * XS_STRIDE];

    const int tid  = threadIdx.x;
    const int wave = tid >> 5;
    const int lane = tid & 31;
    const int rowBase = blockIdx.x << 4;          // n_nodes is a multiple of 16

    // Stage the 16x128 A-tile into LDS (optionally fused ReLU).
    // 2048 floats = 512 float4; 256 threads x 2 float4.
    #pragma unroll
    for (int i = 0; i < 2; ++i) {
        int idx = tid + i * 256;                  // float4 index
        int r   = idx >> 5;                       // 32 float4 per row
        int c4  = idx & 31;
        float4 v = reinterpret_cast<const float4*>(
                       X + (size_t)(rowBase + r) * NFEAT)[c4];
        if (relu_in) {
            v.x = fmaxf(v.x, 0.f); v.y = fmaxf(v.y, 0.f);
            v.z = fmaxf(v.z, 0.f); v.w = fmaxf(v.w, 0.f);
        }
        *reinterpret_cast<float4*>(&Xs[r * XS_STRIDE + c4 * 4]) = v;
    }
    __syncthreads();

    const int colBase = wave << 4;
    const int mn    = lane & 15;                  // M for A, N for B/D
    const int khalf = (lane >> 4) << 1;           // 0 (lanes 0-15) or 2 (16-31)

    v8f acc = {0.f, 0.f, 0.f, 0.f, 0.f, 0.f, 0.f, 0.f};

    #pragma unroll 4
    for (int k0 = 0; k0 < NFEAT; k0 += 4) {
        const int k = k0 + khalf;
        // A fragment (16x4 f32): vgpr0 = K=k, vgpr1 = K=k+1, row M = lane&15
        v2f a;
        a.x = Xs[mn * XS_STRIDE + k];
        a.y = Xs[mn * XS_STRIDE + k + 1];
        // B fragment (4x16 f32): vgpr0 = row K=k, vgpr1 = row K=k+1, col N
        v2f b;
        b.x = W[(size_t)k       * NFEAT + colBase + mn];
        b.y = W[(size_t)(k + 1) * NFEAT + colBase + mn];
        acc = __builtin_amdgcn_wmma_f32_16x16x4_f32(
            /*neg_a=*/false, a, /*neg_b=*/false, b,
            /*c_mod=*/(short)0, acc, /*reuse_a=*/false, /*reuse_b=*/false);
    }

    // D (16x16 f32): vgpr v -> M = (lane>=16 ? 8 : 0) + v, N = lane&15
    const int rowOff = (lane >> 4) << 3;
    float* outp = H + (size_t)(rowBase + rowOff) * NFEAT + colBase + mn;
    #pragma unroll
    for (int v = 0; v < 8; ++v) outp[(size_t)v * NFEAT] = acc[v];
}

// ---------------------------------------------------------------------------
// Per-node attention scalars: s_src[i] = h[i].a_src, s_dst[i] = h[i].a_dst
// One wave per node, float4 loads, xor-shuffle reduction (wave32).
// ---------------------------------------------------------------------------
__global__ __launch_bounds__(256) void gat_node_scores(
    const float* __restrict__ H, const float* __restrict__ aw,
    float* __restrict__ s_src, float* __restrict__ s_dst, int n_nodes)
{
    const int wave = threadIdx.x >> 5;
    const int lane = threadIdx.x & 31;
    const int node = blockIdx.x * 8 + wave;
    if (node >= n_nodes) return;

    const float4 h4 = reinterpret_cast<const float4*>(H + (size_t)node * NFEAT)[lane];
    const float4 as = reinterpret_cast<const float4*>(aw)[lane];
    const float4 ad = reinterpret_cast<const float4*>(aw + NFEAT)[lane];

    float ps = h4.x * as.x + h4.y * as.y + h4.z * as.z + h4.w * as.w;
    float pd = h4.x * ad.x + h4.y * ad.y + h4.z * ad.z + h4.w * ad.w;
    #pragma unroll
    for (int off = 16; off > 0; off >>= 1) {
        ps += __shfl_xor(ps, off, 32);
        pd += __shfl_xor(pd, off, 32);
    }
    if (lane == 0) { s_src[node] = ps; s_dst[node] = pd; }
}

__global__ void gat_fill_zero(float* __restrict__ p, int n)
{
    int i = blockIdx.x * blockDim.x + threadIdx.x;
    if (i < n) p[i] = 0.f;
}

// ---------------------------------------------------------------------------
// Per-edge: e = leakyrelu(s_src[src] + s_dst[dst] + b); w = exp(e);
// wbuf[i] = w; denom[src] += w (atomic).
// ---------------------------------------------------------------------------
__global__ void gat_edge_w(
    const int* __restrict__ src, const int* __restrict__ dst,
    const float* __restrict__ s_src, const float* __restrict__ s_dst,
    const float* __restrict__ bias,
    float* __restrict__ wbuf, float* __restrict__ denom, int n_edges)
{
    int i = blockIdx.x * blockDim.x + threadIdx.x;
    if (i >= n_edges) return;
    const int s = src[i], d = dst[i];
    float e = s_src[s] + s_dst[d] + bias[0];
    e = (e >= 0.f) ? e : 0.05f * e;
    const float w = __expf(e);
    wbuf[i] = w;
    atomicAdd(&denom[s], w);
}

// ---------------------------------------------------------------------------
// Per-edge aggregation: alpha = w/denom[src]; out[src] += alpha * h[dst].
// One wave per edge: float4 gather of h[dst], 4 f32 atomics per lane.
// ---------------------------------------------------------------------------
__global__ __launch_bounds__(256) void gat_edge_agg(
    const int* __restrict__ src, const int* __restrict__ dst,
    const float* __restrict__ wbuf, const float* __restrict__ denom,
    const float* __restrict__ H, float* __restrict__ out,
    float* __restrict__ alpha_out, int n_edges)
{
    const int wave = threadIdx.x >> 5;
    const int lane = threadIdx.x & 31;
    const int e = blockIdx.x * 8 + wave;
    if (e >= n_edges) return;

    const int s = src[e], d = dst[e];
    const float alpha = wbuf[e] / denom[s];
    if (lane == 0) alpha_out[e] = alpha;

    const float4 h4 = reinterpret_cast<const float4*>(H + (size_t)d * NFEAT)[lane];
    float* o = out + (size_t)s * NFEAT + lane * 4;
    atomicAdd(o + 0, alpha * h4.x);
    atomicAdd(o + 1, alpha * h4.y);
    atomicAdd(o + 2, alpha * h4.z);
    atomicAdd(o + 3, alpha * h4.w);
}

// ---------------------------------------------------------------------------
// logits = relu(h) @ fcW + fcb ; out = log_softmax(logits). One thread/node,
// fcW (128x16) staged in LDS.
// ---------------------------------------------------------------------------
__global__ __launch_bounds__(256) void gat_fc_logsoftmax(
    const float* __restrict__ H, const float* __restrict__ fcW,
    const float* __restrict__ fcb, float* __restrict__ out, int n_nodes)
{
    __shared__ float Ws[NFEAT * 16];
    __shared__ float bs[16];
    for (int i = threadIdx.x; i < NFEAT * 16; i += 256) Ws[i] = fcW[i];
    if (threadIdx.x < 16) bs[threadIdx.x] = fcb[threadIdx.x];
    __syncthreads();

    const int node = blockIdx.x * 256 + threadIdx.x;
    if (node >= n_nodes) return;

    float lg[16];
    #pragma unroll
    for (int c = 0; c < 16; ++c) lg[c] = bs[c];

    const float* hp = H + (size_t)node * NFEAT;
    for (int k = 0; k < NFEAT; ++k) {
        const float hk = fmaxf(hp[k], 0.f);
        #pragma unroll
        for (int c = 0; c < 16; ++c) lg[c] += hk * Ws[k * 16 + c];
    }

    float m = lg[0];
    #pragma unroll
    for (int c = 1; c < 16; ++c) m = fmaxf(m, lg[c]);
    float ssum = 0.f;
    #pragma unroll
    for (int c = 0; c < 16; ++c) ssum += __expf(lg[c] - m);
    const float lse = m + __logf(ssum);
    #pragma unroll
    for (int c = 0; c < 16; ++c) out[(size_t)node * 16 + c] = lg[c] - lse;
}

// ---------------------------------------------------------------------------
extern "C" void kernel_launch(void* const* d_in, const int* in_sizes, int n_in,
                              void* d_out, int out_size, void* d_ws, size_t ws_size,
                              hipStream_t stream)
{
    const float* x   = (const float*)d_in[0];
    const int*   src = (const int*)d_in[1];
    const int*   dst = (const int*)d_in[2];
    const float* W1  = (const float*)d_in[3];
    const float* a1w = (const float*)d_in[4];
    const float* a1b = (const float*)d_in[5];
    const float* W2  = (const float*)d_in[6];
    const float* a2w = (const float*)d_in[7];
    const float* a2b = (const float*)d_in[8];
    const float* fcW = (const float*)d_in[9];
    const float* fcb = (const float*)d_in[10];
    float* out = (float*)d_out;

    const int n_nodes = in_sizes[0] / NFEAT;   // 50000
    const int n_edges = in_sizes[1];           // 800000

    // Workspace layout (buffers reused across layers)
    char* ws = (char*)d_ws;
    const size_t featBytes = (size_t)n_nodes * NFEAT * sizeof(float);
    float* bufH  = (float*)ws; ws += featBytes;                     // gemm out (h1, then h2)
    float* bufA  = (float*)ws; ws += featBytes;                     // aggregated out (agg1, then agg2)
    float* wbuf  = (float*)ws; ws += (size_t)n_edges * sizeof(float);
    float* ssrc  = (float*)ws; ws += (size_t)n_nodes * sizeof(float);
    float* sdst  = (float*)ws; ws += (size_t)n_nodes * sizeof(float);
    float* denom = (float*)ws;

    const dim3 blk(256);
    const int gGemm  = n_nodes / 16;                  // 3125
    const int gNodeW = (n_nodes + 7) / 8;             // wave-per-node
    const int gNodeT = (n_nodes + 255) / 256;
    const int gFeat  = (n_nodes * NFEAT + 255) / 256;
    const int gEdgeT = (n_edges + 255) / 256;
    const int gEdgeW = (n_edges + 7) / 8;             // wave-per-edge

    float* alpha2 = out + (size_t)n_nodes * 16;       // tuple output #2

    // ---- Layer 1 ----
    gat_gemm128<<<gGemm, blk, 0, stream>>>(x, W1, bufH, 0);
    gat_node_scores<<<gNodeW, blk, 0, stream>>>(bufH, a1w, ssrc, sdst, n_nodes);
    gat_fill_zero<<<gNodeT, blk, 0, stream>>>(denom, n_nodes);
    gat_fill_zero<<<gFeat, blk, 0, stream>>>(bufA, n_nodes * NFEAT);
    gat_edge_w<<<gEdgeT, blk, 0, stream>>>(src, dst, ssrc, sdst, a1b, wbuf, denom, n_edges);
    gat_edge_agg<<<gEdgeW, blk, 0, stream>>>(src, dst, wbuf, denom, bufH, bufA,
                                             /*alpha discarded:*/ wbuf, n_edges);

    // ---- Layer 2 (ReLU fused into GEMM A-load) ----
    gat_gemm128<<<gGemm, blk, 0, stream>>>(bufA, W2, bufH, 1);
    gat_node_scores<<<gNodeW, blk, 0, stream>>>(bufH, a2w, ssrc, sdst, n_nodes);
    gat_fill_zero<<<gNodeT, blk, 0, stream>>>(denom, n_nodes);
    gat_fill_zero<<<gFeat, blk, 0, stream>>>(bufA, n_nodes * NFEAT);
    gat_edge_w<<<gEdgeT, blk, 0, stream>>>(src, dst, ssrc, sdst, a2b, wbuf, denom, n_edges);
    gat_edge_agg<<<gEdgeW, blk, 0, stream>>>(src, dst, wbuf, denom, bufH, bufA,
                                             alpha2, n_edges);

    // ---- FC + log_softmax (ReLU fused) ----
    gat_fc_logsoftmax<<<gNodeT, blk, 0, stream>>>(bufA, fcW, fcb, out, n_nodes);
}